// PyGGCNModel_48893907698330
// MI455X (gfx1250) — compile-verified
//
#include <hip/hip_runtime.h>
#include <hip/hip_bf16.h>

typedef __attribute__((ext_vector_type(2))) float v2f;
typedef __attribute__((ext_vector_type(8))) float v8f;

// ---------------------------------------------------------------- utilities

__global__ void fill_kernel(float* __restrict__ p, long long n, float v) {
    long long i = blockIdx.x * (long long)blockDim.x + threadIdx.x;
    long long s = gridDim.x * (long long)blockDim.x;
    for (; i < n; i += s) p[i] = v;
}

__global__ void deg_kernel(const long long* __restrict__ dst, float* __restrict__ deg,
                           long long E) {
    long long i = blockIdx.x * (long long)blockDim.x + threadIdx.x;
    long long s = gridDim.x * (long long)blockDim.x;
    for (; i < E; i += s) {
        unsafeAtomicAdd(&deg[dst[i]], 1.0f);
    }
}

__global__ void dinv_kernel(const float* __restrict__ deg, float* __restrict__ dinv,
                            long long n) {
    long long i = blockIdx.x * (long long)blockDim.x + threadIdx.x;
    long long s = gridDim.x * (long long)blockDim.x;
    for (; i < n; i += s) dinv[i] = rsqrtf(deg[i]);   // deg >= 1 due to self-loop
}

// -------------------------------------------------- GEMM1: [N,128] x [128,64]
// One wave computes one 16x16 tile of H via V_WMMA_F32_16X16X4_F32.
// Block = 256 threads = 8 waves = 2 m-tiles x 4 n-tiles -> 32 rows x 64 cols.
// Row index is CLAMPED (not branch-guarded) so the inner loop carries no EXEC
// manipulation: pure global_load_b64 + ds_load_2addr + v_wmma stream.

__global__ __launch_bounds__(256) void gemm1_wmma(const float* __restrict__ X,
                                                  const float* __restrict__ W,
                                                  float* __restrict__ H, int N) {
    __shared__ float Wlds[128 * 64];
    int tid = threadIdx.x;
    for (int i = tid; i < 128 * 64; i += 256) Wlds[i] = W[i];
    __syncthreads();

    int wave = tid >> 5, lane = tid & 31;
    int msub = wave >> 2, nsub = wave & 3;
    int row_base = blockIdx.x * 32 + msub * 16;
    int n_base = nsub * 16;
    int lrow = lane & 15;
    int khalf = lane >> 4;              // lanes 16..31 carry K+2, K+3

    int rowA = row_base + lrow;
    if (rowA >= N) rowA = N - 1;        // clamp; phantom rows discarded at store
    const float* xrow = X + (size_t)rowA * 128 + khalf * 2;
    int nB = n_base + lrow;

    v8f acc = {0.f, 0.f, 0.f, 0.f, 0.f, 0.f, 0.f, 0.f};
#pragma unroll
    for (int k = 0; k < 128; k += 4) {
        v2f a = *(const v2f*)(xrow + k);      // 8B-aligned -> global_load_b64
        int kB = k + khalf * 2;
        v2f b;
        b.x = Wlds[kB * 64 + nB];
        b.y = Wlds[(kB + 1) * 64 + nB];       // stride 64 -> ds_load_2addr_stride64
        acc = __builtin_amdgcn_wmma_f32_16x16x4_f32(false, a, false, b,
                                                    (short)0, acc, false, false);
    }

    int col = n_base + lrow;
    int row0 = row_base + khalf * 8;    // C: VGPR i -> M=i (lanes<16) / M=8+i
#pragma unroll
    for (int i = 0; i < 8; ++i) {
        int r = row0 + i;
        if (r < N) H[(size_t)r * 64 + col] = acc[i];
    }
}

// -------------------------------------------------- GEMM2: [N,64] x [64,40]
// N padded to 48 inside LDS; 192 threads = 6 waves = 2 m-tiles x 3 n-tiles.

__global__ __launch_bounds__(192) void gemm2_wmma(const float* __restrict__ Hin,
                                                  const float* __restrict__ W,
                                                  float* __restrict__ Hout,
                                                  int N, int OUT_F) {
    __shared__ float Wlds[64 * 48];
    int tid = threadIdx.x;
    for (int i = tid; i < 64 * 48; i += 192) {
        int k = i / 48, n = i - k * 48;
        Wlds[i] = (n < OUT_F) ? W[k * OUT_F + n] : 0.f;
    }
    __syncthreads();

    int wave = tid >> 5, lane = tid & 31;
    int msub = wave / 3, nsub = wave - msub * 3;
    int row_base = blockIdx.x * 32 + msub * 16;
    int n_base = nsub * 16;
    int lrow = lane & 15;
    int khalf = lane >> 4;

    int rowA = row_base + lrow;
    if (rowA >= N) rowA = N - 1;        // clamp; stores are guarded below
    const float* hrow = Hin + (size_t)rowA * 64 + khalf * 2;
    int nB = n_base + lrow;

    v8f acc = {0.f, 0.f, 0.f, 0.f, 0.f, 0.f, 0.f, 0.f};
#pragma unroll
    for (int k = 0; k < 64; k += 4) {
        v2f a = *(const v2f*)(hrow + k);
        int kB = k + khalf * 2;
        v2f b;
        b.x = Wlds[kB * 48 + nB];
        b.y = Wlds[(kB + 1) * 48 + nB];
        acc = __builtin_amdgcn_wmma_f32_16x16x4_f32(false, a, false, b,
                                                    (short)0, acc, false, false);
    }

    int col = n_base + lrow;
    if (col < OUT_F) {
        int row0 = row_base + khalf * 8;
#pragma unroll
        for (int i = 0; i < 8; ++i) {
            int r = row0 + i;
            if (r < N) Hout[(size_t)r * OUT_F + col] = acc[i];
        }
    }
}

// ------------------------------------------- edge scatter: out[dst] += w*h[src]
// One wave per edge iteration; lanes stripe the feature dimension (coalesced).
// Prefetch the next iteration's source row into cache (global_prefetch_b8).

__global__ __launch_bounds__(256) void scatter_kernel(const long long* __restrict__ src,
                                                      const long long* __restrict__ dst,
                                                      const float* __restrict__ dinv,
                                                      const float* __restrict__ H,
                                                      float* __restrict__ out,
                                                      int F, long long E) {
    int lane = threadIdx.x & 31;
    long long wave   = (blockIdx.x * (long long)blockDim.x + threadIdx.x) >> 5;
    long long stride = (gridDim.x * (long long)blockDim.x) >> 5;
    for (long long e = wave; e < E; e += stride) {
        long long s = src[e];
        long long d = dst[e];
        long long en = e + stride;
        long long sn = (en < E) ? src[en] : s;
        __builtin_prefetch(&H[sn * F], 0, 0);
        float w = dinv[s] * dinv[d];
        const float* hs = H + s * (long long)F;
        float* od = out + d * (long long)F;
        for (int f = lane; f < F; f += 32) {
            unsafeAtomicAdd(&od[f], w * hs[f]);
        }
    }
}

// ------------------- fused: out = act(agg + dinv^2 * hlin + bias) (elementwise)
// Wave-per-row, lane-per-feature: no integer division, fully coalesced.

__global__ __launch_bounds__(256) void combine_kernel(const float* agg,
                                                      const float* hlin, float* outp,
                                                      const float* __restrict__ dinv,
                                                      const float* __restrict__ bias,
                                                      int F, long long N, int relu) {
    int lane = threadIdx.x & 31;
    long long wave   = (blockIdx.x * (long long)blockDim.x + threadIdx.x) >> 5;
    long long stride = (gridDim.x * (long long)blockDim.x) >> 5;
    for (long long row = wave; row < N; row += stride) {
        float di = dinv[row];
        float w = di * di;
        long long base = row * (long long)F;
        for (int f = lane; f < F; f += 32) {
            float v = agg[base + f] + w * hlin[base + f] + bias[f];
            outp[base + f] = relu ? fmaxf(v, 0.f) : v;
        }
    }
}

// ---------------------------------------------------------------------- launch

extern "C" void kernel_launch(void* const* d_in, const int* in_sizes, int n_in,
                              void* d_out, int out_size, void* d_ws, size_t ws_size,
                              hipStream_t stream) {
    const float*     x  = (const float*)d_in[0];
    const long long* ei = (const long long*)d_in[1];
    const float*     W1 = (const float*)d_in[2];
    const float*     b1 = (const float*)d_in[3];
    const float*     W2 = (const float*)d_in[4];
    const float*     b2 = (const float*)d_in[5];
    float* out = (float*)d_out;

    const int HID   = in_sizes[3];            // 64
    const int OUT_F = in_sizes[5];            // 40
    const int IN_F  = in_sizes[2] / HID;      // 128
    const long long N = in_sizes[0] / IN_F;   // 100000
    const long long E = in_sizes[1] / 2;      // 1600000

    const long long* src = ei;
    const long long* dst = ei + E;

    char* ws = (char*)d_ws;
    float* deg   = (float*)ws; ws += sizeof(float) * N;
    float* dinv  = (float*)ws; ws += sizeof(float) * N;
    float* hlin1 = (float*)ws; ws += sizeof(float) * N * HID;   // reused as h1 in place
    float* agg1  = (float*)ws; ws += sizeof(float) * N * HID;
    float* hlin2 = (float*)ws; ws += sizeof(float) * N * OUT_F;

    const int T = 256;
    int nblk    = (int)((N + T - 1) / T);
    int eblk    = (int)((E + T - 1) / T);
    int gemmblk = (int)((N + 31) / 32);
    int sctblk  = 8192;   // ~65k waves, a few edges each
    int cmbblk  = 4096;   // ~32k waves over 100k rows

    // degrees (self-loop counts as 1) and symmetric norm
    fill_kernel<<<nblk, T, 0, stream>>>(deg, N, 1.0f);
    deg_kernel<<<eblk, T, 0, stream>>>(dst, deg, E);
    dinv_kernel<<<nblk, T, 0, stream>>>(deg, dinv, N);

    // layer 1
    gemm1_wmma<<<gemmblk, 256, 0, stream>>>(x, W1, hlin1, (int)N);
    fill_kernel<<<sctblk, T, 0, stream>>>(agg1, N * HID, 0.0f);
    scatter_kernel<<<sctblk, T, 0, stream>>>(src, dst, dinv, hlin1, agg1, HID, E);
    combine_kernel<<<cmbblk, T, 0, stream>>>(agg1, hlin1, hlin1, dinv, b1, HID,
                                             N, 1);    // + self + bias + relu

    // layer 2
    gemm2_wmma<<<gemmblk, 192, 0, stream>>>(hlin1, W2, hlin2, (int)N, OUT_F);
    fill_kernel<<<sctblk, T, 0, stream>>>(out, N * OUT_F, 0.0f);
    scatter_kernel<<<sctblk, T, 0, stream>>>(src, dst, dinv, hlin2, out, OUT_F, E);
    combine_kernel<<<cmbblk, T, 0, stream>>>(out, hlin2, out, dinv, b2, OUT_F,
                                             N, 0);    // + self + bias
}